// CombinedLoss_773094113606
// MI455X (gfx1250) — compile-verified
//
#include <hip/hip_runtime.h>
#include <hip/hip_bf16.h>
#include <stdint.h>

// ---------------------------------------------------------------------------
// CombinedLoss (BCE + Dice + Boundary) for (32,1,768,768) f32 on MI455X.
// Single fused bandwidth-bound pass:
//   - target tile (<=18 x 768 f32, incl. 3x3 halo) staged into LDS by the
//     CDNA5 Tensor Data Mover (tensor_load_to_lds + s_wait_tensorcnt)
//   - pred streamed with coalesced b128 global loads, software-pipelined one
//     group ahead; first group issued BEFORE the TDM wait/barrier
//   - 3x3 dilate/erode from LDS with clamp-replicate padding (identical to
//     SAME +/-inf padding for max/min windows): 3x(b128 + 2x b32) per group
//   - reduction: regs -> wave32 shfl (ds_permute) -> LDS -> per-block
//     partials in d_ws -> tiny deterministic f64 finalize kernel
// ---------------------------------------------------------------------------

#define B_        32
#define H_        768
#define W_        768
#define BAND_ROWS 16
#define BANDS     (H_ / BAND_ROWS)   // 48 bands per image
#define NBLOCKS   (B_ * BANDS)       // 1536 workgroups
#define TILE_ROWS (BAND_ROWS + 2)    // 18 rows incl. halo
#define NPART     5                  // bce, sum_p, sum_t, sum_pt, bce*bmask
#define NGROUPS   12                 // 12 float4 groups -> 48 px / thread

typedef unsigned int u32x4 __attribute__((ext_vector_type(4)));
typedef int          i32x8 __attribute__((ext_vector_type(8)));
typedef int          i32x4 __attribute__((ext_vector_type(4)));

#if __has_include(<hip/amd_detail/amd_gfx1250_TDM.h>)
#define TDM_ARITY6 1   // therock-10.0 headers -> 6-arg builtin
#endif

__device__ __forceinline__ float max3f(float a, float b, float c) {
  return fmaxf(fmaxf(a, b), c);
}
__device__ __forceinline__ float min3f(float a, float b, float c) {
  return fminf(fminf(a, b), c);
}

__global__ __launch_bounds__(256)
void combined_loss_main(const float* __restrict__ pred,
                        const float* __restrict__ target,
                        float* __restrict__ partials) {
  __shared__ __align__(16) float st[TILE_ROWS * W_];  // 55,296 B (<< 320 KB)
  __shared__ float wred[8][NPART];                    // cross-wave reduction

  const int blk   = blockIdx.x;
  const int img   = blk / BANDS;
  const int band  = blk % BANDS;
  const int r0    = band * BAND_ROWS;
  const int rlo   = (r0 == 0) ? 0 : (r0 - 1);
  const int rhi   = (r0 + BAND_ROWS > H_ - 1) ? (H_ - 1) : (r0 + BAND_ROWS);
  const int nrows = rhi - rlo + 1;         // 17 at image edges, 18 interior
  const int tid   = threadIdx.x;

#if __has_builtin(__builtin_amdgcn_tensor_load_to_lds)
  // --- TDM: async DMA of the (nrows x 768) f32 tile of `target` into LDS ---
  if (tid < 32) {  // wave 0 issues; TENSORcnt is per-wave
    const uint64_t gaddr =
        (uint64_t)(uintptr_t)(target + ((size_t)img * H_ + (size_t)rlo) * W_);
    // Low 32 bits of the flat address of a __shared__ object == LDS offset.
    const uint32_t lds_byte = (uint32_t)(uintptr_t)(&st[0]);

    // D# group 0 (ISA async_tensor Sec 8.3): count=1 | lds_addr |
    // global_addr[56:0] | type=2 ("image") in bits [127:126].
    u32x4 g0;
    g0.x = 1u;
    g0.y = lds_byte;
    g0.z = (uint32_t)gaddr;
    g0.w = ((uint32_t)(gaddr >> 32) & 0x01FFFFFFu) | 0x80000000u;

    // D# group 1 (Sec 8.4): wg_mask=0, data_size=2 (4B); tensor 768 x nrows,
    // tile 768 x nrows, tile_dim2=0, tensor_dim0_stride=768.
    i32x8 g1;
    g1.s0 = (int)(2u << 16);                  // data_size = 4 bytes
    g1.s1 = (int)((uint32_t)W_ << 16);        // tensor_dim0[15:0]
    g1.s2 = (int)((uint32_t)nrows << 16);     // tensor_dim0 hi=0 | tensor_dim1 lo
    g1.s3 = (int)((uint32_t)W_ << 16);        // tensor_dim1 hi=0 | tile_dim0
    g1.s4 = nrows;                            // tile_dim1 | tile_dim2=0
    g1.s5 = W_;                               // tensor_dim0_stride[31:0]
    g1.s6 = 0;
    g1.s7 = 0;

    i32x4 gz4 = {0, 0, 0, 0};                 // 2D tensor: groups 2/3 unused
#if defined(TDM_ARITY6)
    i32x8 gz8 = {0, 0, 0, 0, 0, 0, 0, 0};
    __builtin_amdgcn_tensor_load_to_lds(g0, g1, gz4, gz4, gz8, 0);
#else
    __builtin_amdgcn_tensor_load_to_lds(g0, g1, gz4, gz4, 0);
#endif
  }
#endif

  // --- per-thread geometry; clamp-replicate rows for the 3x3 window --------
  const int rowi   = tid >> 4;            // 0..15 : row within band
  const int lane16 = tid & 15;
  const int r      = r0 + rowi;           // global row
  const int rowm   = (r - rlo) * W_;      // center row offset in LDS
  const int rowt   = (r > 0)      ? rowm - W_ : rowm;   // clamped top
  const int rowb   = (r < H_ - 1) ? rowm + W_ : rowm;   // clamped bottom
  const float* prow = pred + ((size_t)img * H_ + (size_t)r) * W_;

  // Prefetch pred group 0 while the TDM runs.
  float4 pcur = *reinterpret_cast<const float4*>(prow + lane16 * 4);

#if __has_builtin(__builtin_amdgcn_tensor_load_to_lds)
#if __has_builtin(__builtin_amdgcn_s_wait_tensorcnt)
  if (tid < 32) __builtin_amdgcn_s_wait_tensorcnt(0);
#endif
#else
  // Fallback: cooperative load (rows are contiguous in memory).
  for (int i = tid; i < nrows * W_; i += 256) {
    st[i] = target[((size_t)img * H_ + (size_t)rlo) * W_ + i];
  }
#endif
  __syncthreads();

  // --- fused elementwise + 3x3 morphology + partial reductions -------------
  float s_bce = 0.f, s_p = 0.f, s_t = 0.f, s_pt = 0.f, s_bb = 0.f;

#pragma unroll 2
  for (int g = 0; g < NGROUPS; ++g) {
    const int cbase = lane16 * 4 + g * 64;
    const int cl = (cbase == 0)      ? 0       : cbase - 1;  // clamp-replicate
    const int cr = (cbase + 4 == W_) ? W_ - 1  : cbase + 4;

    // 3 rows x (1x ds_load_b128 + 2x ds_load_b32): columns [cbase-1..cbase+4]
    const float4 vm = *reinterpret_cast<const float4*>(&st[rowm + cbase]);
    const float4 vt = *reinterpret_cast<const float4*>(&st[rowt + cbase]);
    const float4 vb = *reinterpret_cast<const float4*>(&st[rowb + cbase]);
    const float lm = st[rowm + cl], lt = st[rowt + cl], lb = st[rowb + cl];
    const float rm = st[rowm + cr], rt = st[rowt + cr], rb = st[rowb + cr];

    // Vertical 3-row max/min per column (branch-free).
    float cmax[6], cmin[6];
    cmax[0] = max3f(lm, lt, lb);       cmin[0] = min3f(lm, lt, lb);
    cmax[1] = max3f(vm.x, vt.x, vb.x); cmin[1] = min3f(vm.x, vt.x, vb.x);
    cmax[2] = max3f(vm.y, vt.y, vb.y); cmin[2] = min3f(vm.y, vt.y, vb.y);
    cmax[3] = max3f(vm.z, vt.z, vb.z); cmin[3] = min3f(vm.z, vt.z, vb.z);
    cmax[4] = max3f(vm.w, vt.w, vb.w); cmin[4] = min3f(vm.w, vt.w, vb.w);
    cmax[5] = max3f(rm, rt, rb);       cmin[5] = min3f(rm, rt, rb);

    // Software pipeline: fetch next pred group while computing this one.
    const int gn = (g + 1 < NGROUPS) ? g + 1 : g;
    const float4 pnext =
        *reinterpret_cast<const float4*>(prow + (lane16 * 4 + gn * 64));

    const float px[4] = {pcur.x, pcur.y, pcur.z, pcur.w};
    const float tc[4] = {vm.x, vm.y, vm.z, vm.w};
#pragma unroll
    for (int k = 0; k < 4; ++k) {
      const float x     = px[k];
      const float t     = tc[k];
      const float bmask = max3f(cmax[k], cmax[k + 1], cmax[k + 2]) -
                          min3f(cmin[k], cmin[k + 1], cmin[k + 2]);

      // Stable BCE-with-logits: bce = log1p(exp(-|x|)) + max(x,0) - t*x
      const float e    = __expf(-fabsf(x));
      const float soft = __logf(1.0f + e);
      const float bce  = soft + fmaxf(x, 0.f) - t * x;
      const float inv  = 1.0f / (1.0f + e);
      const float sig  = (x >= 0.f) ? inv : e * inv;

      s_bce += bce;
      s_p   += sig;
      s_t   += t;
      s_pt  += sig * t;
      s_bb  += bce * bmask;
    }
    pcur = pnext;
  }

  // --- wave32 shuffle reduction, then cross-wave via LDS -------------------
#pragma unroll
  for (int off = 16; off > 0; off >>= 1) {
    s_bce += __shfl_down(s_bce, off, 32);
    s_p   += __shfl_down(s_p,   off, 32);
    s_t   += __shfl_down(s_t,   off, 32);
    s_pt  += __shfl_down(s_pt,  off, 32);
    s_bb  += __shfl_down(s_bb,  off, 32);
  }
  const int wid = tid >> 5;  // 8 waves per block (wave32)
  if ((tid & 31) == 0) {
    wred[wid][0] = s_bce; wred[wid][1] = s_p; wred[wid][2] = s_t;
    wred[wid][3] = s_pt;  wred[wid][4] = s_bb;
  }
  __syncthreads();
  if (tid == 0) {
    float a[NPART] = {0.f, 0.f, 0.f, 0.f, 0.f};
    for (int w = 0; w < 8; ++w)
#pragma unroll
      for (int k = 0; k < NPART; ++k) a[k] += wred[w][k];
    float* po = partials + (size_t)blk * NPART;
#pragma unroll
    for (int k = 0; k < NPART; ++k) po[k] = a[k];
  }
}

// --- deterministic finalize: 1536x5 partials -> (total, bce, dice, boundary)
__global__ __launch_bounds__(256)
void combined_loss_finalize(const float* __restrict__ partials,
                            float* __restrict__ out) {
  __shared__ double red[256][NPART];
  double s[NPART] = {0.0, 0.0, 0.0, 0.0, 0.0};
  for (int b = threadIdx.x; b < NBLOCKS; b += 256) {
#pragma unroll
    for (int k = 0; k < NPART; ++k) s[k] += (double)partials[b * NPART + k];
  }
#pragma unroll
  for (int k = 0; k < NPART; ++k) red[threadIdx.x][k] = s[k];
  __syncthreads();
  for (int stride = 128; stride > 0; stride >>= 1) {
    if ((int)threadIdx.x < stride)
#pragma unroll
      for (int k = 0; k < NPART; ++k)
        red[threadIdx.x][k] += red[threadIdx.x + stride][k];
    __syncthreads();
  }
  if (threadIdx.x == 0) {
    const double sum_bce = red[0][0], sum_p = red[0][1], sum_t = red[0][2],
                 sum_pt = red[0][3], sum_bb = red[0][4];
    const double invN = 1.0 / (double)((size_t)B_ * H_ * W_);
    const double bce      = sum_bce * invN;
    const double dice     = 1.0 - (2.0 * sum_pt + 1.0) / (sum_p + sum_t + 1.0);
    const double boundary = (sum_bce + 9.0 * sum_bb) * invN;  // mean(bce*(1+9m))
    const double total    = bce + dice + boundary;
    out[0] = (float)total;
    out[1] = (float)bce;
    out[2] = (float)dice;
    out[3] = (float)boundary;
  }
}

extern "C" void kernel_launch(void* const* d_in, const int* in_sizes, int n_in,
                              void* d_out, int out_size, void* d_ws, size_t ws_size,
                              hipStream_t stream) {
  const float* pred   = (const float*)d_in[0];
  const float* target = (const float*)d_in[1];
  float* partials     = (float*)d_ws;   // NBLOCKS * NPART * 4 = 30 KB scratch

  combined_loss_main<<<NBLOCKS, 256, 0, stream>>>(pred, target, partials);
  combined_loss_finalize<<<1, 256, 0, stream>>>(partials, (float*)d_out);
}